// EMCriterion_29807073034918
// MI455X (gfx1250) — compile-verified
//
#include <hip/hip_runtime.h>
#include <stdint.h>
#include <math.h>

// Problem constants (from reference)
#define B_ 4
#define H_ 256
#define W_ 256
#define Q_ 128
#define E_ 64
#define N_ 64
#define HW_ (H_ * W_)
#define NO_W 0.1f
#define HUB_D 0.1f

#define BLOCKS_PER_BATCH 1024
#define WAVES_PER_BLOCK 8
#define THREADS (WAVES_PER_BLOCK * 32)
// ws layout: float[ gridBlocks * 4 ] : {bce_union_sum, union_sum, num_b, den_b}
// gridBlocks = B_ * BLOCKS_PER_BATCH = 4096  -> 64 KiB of workspace

__device__ __forceinline__ float softplusf(float x) {
    // numerically stable softplus, matches jax.nn.softplus in f32
    return fmaxf(x, 0.0f) + log1pf(expf(-fabsf(x)));
}

__device__ __forceinline__ float wave_sum(float v) {
#pragma unroll
    for (int m = 16; m >= 1; m >>= 1) v += __shfl_xor(v, m, 32);
    return v;
}
__device__ __forceinline__ float wave_max(float v) {
#pragma unroll
    for (int m = 16; m >= 1; m >>= 1) v = fmaxf(v, __shfl_xor(v, m, 32));
    return v;
}

__global__ __launch_bounds__(THREADS) void em_main(
    const float* __restrict__ P,            // pred_seg_logits (B,H,W,Q)
    const float* __restrict__ T,            // true_seg        (B,H,W,E)
    const unsigned char* __restrict__ Pm,   // pred_seg_mask   (B,H,W,Q) bool
    const unsigned char* __restrict__ Tm,   // true_seg_mask   (B,H,W,E) bool
    const int* __restrict__ MI,             // matched_indices (B,2,N) int32
    float* __restrict__ ws)                 // per-block partials
{
    __shared__ int ch0[N_], ch1[N_];
    __shared__ __align__(16) float spbuf[WAVES_PER_BLOCK][Q_];   // 4 KiB
    __shared__ __align__(16) float stbuf[WAVES_PER_BLOCK][E_];   // 2 KiB
    __shared__ __align__(16) unsigned char spmbuf[WAVES_PER_BLOCK][Q_]; // 1 KiB
    __shared__ float wred[WAVES_PER_BLOCK][5];

    const int b    = blockIdx.x / BLOCKS_PER_BATCH;
    const int blk  = blockIdx.x % BLOCKS_PER_BATCH;
    const int tid  = threadIdx.x;
    const int wave = tid >> 5;
    const int lane = tid & 31;

    if (tid < 2 * N_) {
        int v = MI[b * 2 * N_ + tid];   // [b,0,:]=idx0 then [b,1,:]=idx1
        if (tid < N_) ch0[tid] = v; else ch1[tid - N_] = v;
    }
    __syncthreads();

    // each lane owns gathered channels 2*lane, 2*lane+1
    const int c0 = ch0[2 * lane], c1 = ch0[2 * lane + 1];
    const int e0 = ch1[2 * lane], e1 = ch1[2 * lane + 1];

    // LDS byte addresses for the async copies (low 32 bits of generic ptr = LDS offset)
    const unsigned lds_sp  = (unsigned)(uintptr_t)&spbuf[wave][lane * 4];
    const unsigned lds_st  = (unsigned)(uintptr_t)&stbuf[wave][lane * 2];
    const unsigned lds_spm = (unsigned)(uintptr_t)&spmbuf[wave][lane * 4];
    const unsigned va16 = (unsigned)(lane * 16);
    const unsigned va8  = (unsigned)(lane * 8);
    const unsigned va4  = (unsigned)(lane * 4);

    float acc_bceu = 0.f, acc_un = 0.f, acc_tv = 0.f, acc_num = 0.f, acc_den = 0.f;

    for (int pix = blk * WAVES_PER_BLOCK + wave; pix < HW_;
         pix += BLOCKS_PER_BATCH * WAVES_PER_BLOCK) {
        const size_t gp = (size_t)b * HW_ + (size_t)pix;
        const uint64_t prow  = (uint64_t)(uintptr_t)(P  + gp * Q_);  // 512 B row
        const uint64_t trow  = (uint64_t)(uintptr_t)(T  + gp * E_);  // 256 B row
        const uint64_t pmrow = (uint64_t)(uintptr_t)(Pm + gp * Q_);  // 128 B row

        // make sure previous iteration's LDS reads are done before overwrite
        asm volatile("s_wait_dscnt 0" ::: "memory");
        // fully-coalesced wide streams into per-wave LDS staging buffers
        asm volatile("global_load_async_to_lds_b128 %0, %1, %2"
                     :: "v"(lds_sp),  "v"(va16), "s"(prow)  : "memory");
        asm volatile("global_load_async_to_lds_b64 %0, %1, %2"
                     :: "v"(lds_st),  "v"(va8),  "s"(trow)  : "memory");
        asm volatile("global_load_async_to_lds_b32 %0, %1, %2"
                     :: "v"(lds_spm), "v"(va4),  "s"(pmrow) : "memory");

        // true-mask row is only 64 B: direct (naturally coalesced) byte loads
        const unsigned char* tmrow = Tm + gp * E_;
        const float stm0 = tmrow[e0] ? 1.f : 0.f;
        const float stm1 = tmrow[e1] ? 1.f : 0.f;

        asm volatile("s_wait_asynccnt 0" ::: "memory");

        const float sp0 = spbuf[wave][c0], sp1 = spbuf[wave][c1];
        const float st0 = stbuf[wave][e0], st1 = stbuf[wave][e1];
        const float spm0 = spmbuf[wave][c0] ? 1.f : 0.f;
        const float spm1 = spmbuf[wave][c1] ? 1.f : 0.f;

        const float lg0 = sp0 * spm0, lg1 = sp1 * spm1;
        const float tv0 = st0 * stm0, tv1 = st1 * stm1;
        const float un0 = fmaxf(spm0, stm0), un1 = fmaxf(spm1, stm1);
        const float bce0 = softplusf(lg0) - lg0 * tv0;
        const float bce1 = softplusf(lg1) - lg1 * tv1;
        acc_bceu += bce0 * un0 + bce1 * un1;
        acc_un   += un0 + un1;
        acc_tv   += tv0 + tv1;

        // masked softmax over the 64 gathered channels of this pixel
        const float neg0 = spm0 > 0.f ? sp0 : -1e30f;
        const float neg1 = spm1 > 0.f ? sp1 : -1e30f;
        const float m = wave_max(fmaxf(neg0, neg1));
        const float ex0 = expf(neg0 - m), ex1 = expf(neg1 - m);
        const float s = wave_sum(ex0 + ex1);                         // >= 1
        const float a = wave_sum(ex0 * spm0 * tv0 + ex1 * spm1 * tv1);
        const float c = wave_sum(ex0 * spm0 + ex1 * spm1);
        const float inv = 1.0f / s;
        acc_num += 2.0f * a * inv;   // lane-uniform
        acc_den += c * inv;          // lane-uniform
    }

    // per-lane sums -> per-wave totals (num/den already lane-uniform)
    acc_bceu = wave_sum(acc_bceu);
    acc_un   = wave_sum(acc_un);
    acc_tv   = wave_sum(acc_tv);
    if (lane == 0) {
        wred[wave][0] = acc_bceu;
        wred[wave][1] = acc_un;
        wred[wave][2] = acc_tv;
        wred[wave][3] = acc_num;
        wred[wave][4] = acc_den;
    }
    __syncthreads();
    if (tid == 0) {
        float s0 = 0, s1 = 0, s2 = 0, s3 = 0, s4 = 0;
        for (int w = 0; w < WAVES_PER_BLOCK; ++w) {
            s0 += wred[w][0]; s1 += wred[w][1]; s2 += wred[w][2];
            s3 += wred[w][3]; s4 += wred[w][4];
        }
        float* o = ws + (size_t)blockIdx.x * 4;
        o[0] = s0;          // sum(bce * union)
        o[1] = s1;          // sum(union)
        o[2] = s3;          // num_s partial (this batch)
        o[3] = s4 + s2;     // den_s partial: sum(p) + sum(tvals)
    }
}

__global__ __launch_bounds__(256) void em_final(
    const float* __restrict__ logits,   // (B*Q)
    const float* __restrict__ PP,       // (B*Q,2)
    const float* __restrict__ PC,       // (B*Q,2,2)
    const float* __restrict__ TP,       // (B*E,2)
    const int* __restrict__ MI,         // (B,2,N)
    const float* __restrict__ ws,
    float* __restrict__ out)            // 5 floats
{
    __shared__ float red[256];
    __shared__ unsigned char labels[B_ * Q_];
    const int tid = threadIdx.x;

    auto block_sum = [&](float v) -> float {
        red[tid] = v; __syncthreads();
        for (int st = 128; st >= 1; st >>= 1) {
            if (tid < st) red[tid] += red[tid + st];
            __syncthreads();
        }
        float r = red[0]; __syncthreads();
        return r;
    };

    // ---- deterministic reduction of main-kernel partials ----
    float s0 = 0.f, s1 = 0.f;
    float numb[B_], denb[B_];
#pragma unroll
    for (int bb = 0; bb < B_; ++bb) { numb[bb] = 0.f; denb[bb] = 0.f; }
#pragma unroll
    for (int bb = 0; bb < B_; ++bb) {
        for (int i = tid; i < BLOCKS_PER_BATCH; i += 256) {
            const float* p = ws + (size_t)(bb * BLOCKS_PER_BATCH + i) * 4;
            s0 += p[0]; s1 += p[1]; numb[bb] += p[2]; denb[bb] += p[3];
        }
    }
    s0 = block_sum(s0);
    s1 = block_sum(s1);
#pragma unroll
    for (int bb = 0; bb < B_; ++bb) { numb[bb] = block_sum(numb[bb]); denb[bb] = block_sum(denb[bb]); }

    // ---- class loss ----
    labels[tid] = 0; labels[tid + 256] = 0;
    __syncthreads();
    {
        const int b = tid >> 6, n = tid & 63;
        const int i0 = MI[b * 2 * N_ + n];
        labels[b * Q_ + i0] = 1;            // unique per batch -> no conflicts
    }
    __syncthreads();
    float cls = 0.f;
    for (int i = tid; i < B_ * Q_; i += 256) {
        const float x = logits[i];
        const float lab = labels[i] ? 1.f : 0.f;
        const float w = labels[i] ? 1.f : NO_W;
        cls += w * (softplusf(x) - x * lab);
    }

    // ---- NLL + Huber: one matched pair per thread ----
    const int b = tid >> 6, n = tid & 63;
    const int i0 = MI[b * 2 * N_ + n];
    const int i1 = MI[b * 2 * N_ + N_ + n];
    const int qi = b * Q_ + i0;
    const float mu0 = PP[qi * 2 + 0], mu1 = PP[qi * 2 + 1];
    const float L00 = PC[qi * 4 + 0], L10 = PC[qi * 4 + 2], L11 = PC[qi * 4 + 3];
    const int ei = b * E_ + i1;
    const float t0 = TP[ei * 2 + 0], t1 = TP[ei * 2 + 1];
    const float d0 = t0 - mu0, d1 = t1 - mu1;
    const float z0 = d0 / L00;
    const float z1 = (d1 - L10 * z0) / L11;
    float nll = 0.5f * (z0 * z0 + z1 * z1) + logf(L00) + logf(L11)
              + 1.8378770664093453f;                     // log(2*pi)
    nll = fminf(fmaxf(nll, -1e7f), 1e7f);
    const float a0 = fabsf(mu0 - t0), a1 = fabsf(mu1 - t1);
    const float h0 = a0 < HUB_D ? 0.5f * a0 * a0 : HUB_D * (a0 - 0.5f * HUB_D);
    const float h1 = a1 < HUB_D ? 0.5f * a1 * a1 : HUB_D * (a1 - 0.5f * HUB_D);

    const float cls_s = block_sum(cls);
    const float nll_s = block_sum(nll);
    const float hub_s = block_sum(h0 + h1);

    if (tid == 0) {
        out[0] = cls_s / (float)(B_ * Q_);
        out[1] = s0 / s1;
        float dice = 0.f;
        for (int bb = 0; bb < B_; ++bb)
            dice += 1.f - (numb[bb] + 1.f) / (denb[bb] + 1.f);
        out[2] = dice / (float)B_;
        out[3] = nll_s / (float)(B_ * N_);
        out[4] = hub_s / (float)(B_ * N_ * 2);
    }
}

extern "C" void kernel_launch(void* const* d_in, const int* in_sizes, int n_in,
                              void* d_out, int out_size, void* d_ws, size_t ws_size,
                              hipStream_t stream) {
    const float* pred_logits       = (const float*)d_in[0];
    const float* pred_seg_logits   = (const float*)d_in[1];
    const float* true_seg          = (const float*)d_in[2];
    const float* pred_positions    = (const float*)d_in[3];
    const float* pred_chol         = (const float*)d_in[4];
    const float* true_positions    = (const float*)d_in[5];
    const unsigned char* pred_seg_mask = (const unsigned char*)d_in[6];
    const unsigned char* true_seg_mask = (const unsigned char*)d_in[7];
    const int* matched             = (const int*)d_in[8];
    float* ws  = (float*)d_ws;
    float* out = (float*)d_out;

    (void)in_sizes; (void)n_in; (void)out_size; (void)ws_size;

    em_main<<<B_ * BLOCKS_PER_BATCH, THREADS, 0, stream>>>(
        pred_seg_logits, true_seg, pred_seg_mask, true_seg_mask, matched, ws);
    em_final<<<1, 256, 0, stream>>>(
        pred_logits, pred_positions, pred_chol, true_positions, matched, ws, out);
}